// MLP_GAS_35493609734644
// MI455X (gfx1250) — compile-verified
//
#include <hip/hip_runtime.h>
#include <math.h>

typedef float v2f __attribute__((ext_vector_type(2)));
typedef float v8f __attribute__((ext_vector_type(8)));

// ---------------------------------------------------------------------------
// s[d] = sum_{l<L} other[l*D + d]   (float4-vectorized column sum, 15.7MB read)
// ---------------------------------------------------------------------------
__global__ void __launch_bounds__(256)
colsum_kernel(const float* __restrict__ other, float* __restrict__ s,
              int D, int L)
{
    int i = (blockIdx.x * blockDim.x + threadIdx.x) * 4;
    if (i >= D) return;
    float4 acc = *(const float4*)(other + i);
    for (int l = 1; l < L; ++l) {
        float4 v = *(const float4*)(other + (size_t)l * D + i);
        acc.x += v.x; acc.y += v.y; acc.z += v.z; acc.w += v.w;
    }
    *(float4*)(s + i) = acc;
}

// ---------------------------------------------------------------------------
// y += W @ x  via V_WMMA_F32_16X16X4_F32 (f32 matrix pipe, no precision loss).
//
// Per wave: one 16-row block of W, one K-slice of length kchunk.
//   A (16x4 f32, 2 VGPRs/lane):  lane l<16  -> W[m0+l, k+0], W[m0+l, k+1]
//                                lane l>=16 -> W[m0+l-16, k+2], W[m0+l-16, k+3]
//   B (4x16 f32, 2 VGPRs/lane):  every column n carries x[k..k+3] with the
//                                same K split (lanes 0-15 = K0/K1, 16-31 = K2/K3)
// All B columns identical => all D columns equal the partial matvec.
// C/D layout: VGPR r = row m0+r (lanes 0-15) / m0+8+r (lanes 16-31).
// Two accumulators break the WMMA D->C dependence chain; partials are combined
// across K-slices with global_atomic_add_f32 into a zero-initialized y.
// ---------------------------------------------------------------------------
__global__ void __launch_bounds__(256)
matvec_wmma_kernel(const float* __restrict__ W, const float* __restrict__ x,
                   float* __restrict__ y, int M, int K, int kchunk)
{
    const int lane = threadIdx.x & 31;
    const int wave = blockIdx.x * (blockDim.x >> 5) + (threadIdx.x >> 5);
    const int RB   = M >> 4;                 // 16-row blocks
    const int m0   = (wave % RB) << 4;
    const int k0   = (wave / RB) * kchunk;
    const int kh   = lane >> 4;              // 0: K0/K1 half, 1: K2/K3 half
    const int row  = m0 + (lane & 15);

    const float* wp = W + (size_t)row * (size_t)K + k0 + 2 * kh;
    const float* xp = x + k0 + 2 * kh;

    v8f c0 = {};
    v8f c1 = {};
    for (int k = 0; k < kchunk; k += 8) {
        v2f a0 = *(const v2f*)(wp + k);
        v2f b0 = *(const v2f*)(xp + k);
        v2f a1 = *(const v2f*)(wp + k + 4);
        v2f b1 = *(const v2f*)(xp + k + 4);
        c0 = __builtin_amdgcn_wmma_f32_16x16x4_f32(false, a0, false, b0,
                                                   (short)0, c0, false, false);
        c1 = __builtin_amdgcn_wmma_f32_16x16x4_f32(false, a1, false, b1,
                                                   (short)0, c1, false, false);
    }

    if ((lane & 15) == 0) {                  // lane 0 -> rows m0..m0+7, lane 16 -> m0+8..m0+15
        const int mb = m0 + (kh ? 8 : 0);
        #pragma unroll
        for (int r = 0; r < 8; ++r)
            atomicAdd(&y[mb + r], c0[r] + c1[r]);
    }
}

// ---------------------------------------------------------------------------
// Tail: bias+relu, argmax one-hot, fc2 (64x96), fc3 (16x64). One workgroup.
// ---------------------------------------------------------------------------
__global__ void __launch_bounds__(128)
finalize_kernel(const float* __restrict__ y1, const float* __restrict__ agg,
                const float* __restrict__ b11, const float* __restrict__ b12,
                const float* __restrict__ W2, const float* __restrict__ b2,
                const float* __restrict__ W3, const float* __restrict__ b3,
                float* __restrict__ out, int LL)
{
    __shared__ float xbuf[96];
    __shared__ float hbuf[64];
    __shared__ int   amax;
    const int t = threadIdx.x;

    if (t < 64) xbuf[t] = fmaxf(y1[t] + b11[t], 0.0f);
    if (t == 0) {
        float best = -INFINITY; int bi = 0;
        for (int c = 0; c < 32; ++c) {
            float v = agg[c] + (float)LL * b12[c];
            if (v > best) { best = v; bi = c; }   // first-max, matches jnp.argmax
        }
        amax = bi;
    }
    __syncthreads();
    if (t < 32) xbuf[64 + t] = (t == amax) ? 1.0f : 0.0f;  // relu(one-hot)==one-hot
    __syncthreads();
    if (t < 64) {
        float acc = b2[t];
        for (int j = 0; j < 96; ++j) acc = fmaf(W2[t * 96 + j], xbuf[j], acc);
        hbuf[t] = fmaxf(acc, 0.0f);
    }
    __syncthreads();
    if (t < 16) {
        float acc = b3[t];
        for (int j = 0; j < 64; ++j) acc = fmaf(W3[t * 64 + j], hbuf[j], acc);
        out[t] = acc;
    }
}

extern "C" void kernel_launch(void* const* d_in, const int* in_sizes, int n_in,
                              void* d_out, int out_size, void* d_ws, size_t ws_size,
                              hipStream_t stream)
{
    (void)n_in; (void)out_size; (void)ws_size;

    const float* inputs = (const float*)d_in[0];
    const float* other  = (const float*)d_in[1];
    const float* W11    = (const float*)d_in[2];
    const float* b11    = (const float*)d_in[3];
    const float* W12    = (const float*)d_in[4];
    const float* b12    = (const float*)d_in[5];
    const float* W2     = (const float*)d_in[6];
    const float* b2     = (const float*)d_in[7];
    const float* W3     = (const float*)d_in[8];
    const float* b3     = (const float*)d_in[9];
    float* out = (float*)d_out;

    const int D_IN  = in_sizes[0];            // 1048576
    const int D_OTH = in_sizes[4] / 32;       // W12 is [32, D_OTHER] -> 262144
    const int LL    = in_sizes[1] / D_OTH;    // 15

    // Workspace layout: s[D_OTH] | y1[64] | agg[32]
    float* s   = (float*)d_ws;
    float* y1  = s + D_OTH;
    float* agg = y1 + 64;

    // Zero the atomic accumulators (ws is poisoned by the harness).
    hipMemsetAsync(y1, 0, (64 + 32) * sizeof(float), stream);

    // 1) column-sum of other_inputs: 15 x 262144 -> 262144
    {
        const int threads = 256;
        const int blocks  = (D_OTH / 4 + threads - 1) / threads;   // 256
        colsum_kernel<<<blocks, threads, 0, stream>>>(other, s, D_OTH, LL);
    }

    // 2) y1 = W11 @ inputs   (256 MiB stream -> dominate; 4096 waves)
    {
        const int kchunk = 1024;
        const int waves  = (64 / 16) * (D_IN / kchunk);            // 4096
        matvec_wmma_kernel<<<waves * 32 / 256, 256, 0, stream>>>(
            W11, inputs, y1, 64, D_IN, kchunk);
    }

    // 3) agg = W12 @ s       (32 MiB stream; 512 waves)
    {
        const int kchunk = 1024;
        const int waves  = (32 / 16) * (D_OTH / kchunk);           // 512
        matvec_wmma_kernel<<<waves * 32 / 256, 256, 0, stream>>>(
            W12, s, agg, 32, D_OTH, kchunk);
    }

    // 4) bias/relu/argmax/fc2/fc3 tail
    finalize_kernel<<<1, 128, 0, stream>>>(y1, agg, b11, b12, W2, b2, W3, b3,
                                           out, LL);
}